// SkipGram_64037962383539
// MI455X (gfx1250) — compile-verified
//
#include <hip/hip_runtime.h>
#include <hip/hip_bf16.h>

typedef __attribute__((ext_vector_type(2))) float v2f;
typedef __attribute__((ext_vector_type(8))) float v8f;

#define V_ROWS        20000
#define D_DIM         128
#define N_PAIRS       65536
#define COLSUM_BLOCKS 64
#define WAVES_TOTAL   (N_PAIRS / 16)   // 16 pairs per wave -> 4096 waves

// ---------------------------------------------------------------------------
// Phase 1a: per-block partial column sums of E (rows strided by grid).
// blockDim.x == 128 (one thread per column d).
// ---------------------------------------------------------------------------
__global__ void sg_colsum_partial(const float* __restrict__ E,
                                  float* __restrict__ part) {
    const int d = threadIdx.x;
    float acc = 0.0f;
    for (int row = blockIdx.x; row < V_ROWS; row += COLSUM_BLOCKS)
        acc += E[(long)row * D_DIM + d];
    part[blockIdx.x * D_DIM + d] = acc;
}

// Phase 1b: finalize s[d] = sum over blocks of partials (deterministic).
__global__ void sg_colsum_final(const float* __restrict__ part,
                                float* __restrict__ s) {
    const int d = threadIdx.x;
    float acc = 0.0f;
#pragma unroll 8
    for (int b = 0; b < COLSUM_BLOCKS; ++b)
        acc += part[b * D_DIM + d];
    s[d] = acc;
}

// ---------------------------------------------------------------------------
// Phase 2: one wave32 handles 16 pairs via V_WMMA_F32_16X16X4_F32.
//   A (16x4 fp32): M = pair slot, lanes 0-15 hold K={0,1}, lanes 16-31 K={2,3}
//   B (4x16 fp32): N = pair slot, same K split  -> C diag = e_i . e_j
//   B2 = s broadcast into every column          -> C2 diag = e_i . s
// Diagonal ownership (32-bit 16x16 C/D layout):
//   lanes 0..7  : pair p = lane,      value = c[lane]
//   lanes 24..31: pair p = lane - 16, value = c[lane - 24]
// ---------------------------------------------------------------------------
__global__ void sg_pair_cost_wmma(const float* __restrict__ E,
                                  const int* __restrict__ iv,
                                  const int* __restrict__ jv,
                                  const float* __restrict__ s,
                                  float* __restrict__ wavesum) {
    const int gtid  = blockIdx.x * blockDim.x + threadIdx.x;
    const int wave  = gtid >> 5;
    const int lane  = threadIdx.x & 31;
    const int col   = lane & 15;          // pair slot inside the 16-pair tile
    const int khalf = lane >> 4;          // 0 -> K{0,1}, 1 -> K{2,3}

    const int  pair = wave * 16 + col;
    const long irow = (long)iv[pair];
    const long jrow = (long)jv[pair];

    const float* ei = E + irow * D_DIM + khalf * 2;
    const float* ej = E + jrow * D_DIM + khalf * 2;
    const float* sp = s + khalf * 2;

    v8f c  = {};   // accumulates e_i . e_j tile
    v8f c2 = {};   // accumulates e_i . s   tile
#pragma unroll
    for (int kc = 0; kc < D_DIM / 4; ++kc) {
        v2f a  = *(const v2f*)(ei + kc * 4);
        v2f b  = *(const v2f*)(ej + kc * 4);
        v2f bs = *(const v2f*)(sp + kc * 4);
        c  = __builtin_amdgcn_wmma_f32_16x16x4_f32(
                 false, a, false, b,  (short)0, c,  false, false);
        c2 = __builtin_amdgcn_wmma_f32_16x16x4_f32(
                 false, a, false, bs, (short)0, c2, false, false);
    }

    // Extract diagonal (pair dot) and matching row of C2 (e_i . s).
    const bool active = (lane < 8) || (lane >= 24);
    const int  r      = (lane < 8) ? lane : (lane - 24);
    float ad = 0.0f, bd = 0.0f;
#pragma unroll
    for (int q = 0; q < 8; ++q) {
        if (q == r) { ad = c[q]; bd = c2[q]; }
    }

    float cost = 0.0f;
    if (active) {
        // denoms/norms = exp(a)/exp(b) = exp(a - b)
        cost = -__logf(__expf(ad - bd) + 1e-8f);
    }

    // Wave-level reduction (wave32).
#pragma unroll
    for (int off = 16; off > 0; off >>= 1)
        cost += __shfl_xor(cost, off, 32);

    if (lane == 0)
        wavesum[wave] = cost;
}

// ---------------------------------------------------------------------------
// Phase 3: deterministic single-block reduction of the 4096 wave sums.
// ---------------------------------------------------------------------------
__global__ void sg_final_reduce(const float* __restrict__ wavesum,
                                float* __restrict__ out) {
    __shared__ float sm[256];
    float acc = 0.0f;
    for (int i = threadIdx.x; i < WAVES_TOTAL; i += 256)
        acc += wavesum[i];
    sm[threadIdx.x] = acc;
    __syncthreads();
#pragma unroll
    for (int w = 128; w > 0; w >>= 1) {
        if ((int)threadIdx.x < w) sm[threadIdx.x] += sm[threadIdx.x + w];
        __syncthreads();
    }
    if (threadIdx.x == 0)
        out[0] = sm[0] * (1.0f / (float)N_PAIRS);
}

// ---------------------------------------------------------------------------
// Host-side launch. Inputs: [0]=embedding f32 [V*D], [1]=i_vec int, [2]=j_vec.
// ws layout: [0, 64*128) partials | [8192, 8320) s | [8320, 8320+4096) wavesums
// ---------------------------------------------------------------------------
extern "C" void kernel_launch(void* const* d_in, const int* in_sizes, int n_in,
                              void* d_out, int out_size, void* d_ws, size_t ws_size,
                              hipStream_t stream) {
    const float* E  = (const float*)d_in[0];
    const int*   iv = (const int*)d_in[1];
    const int*   jv = (const int*)d_in[2];
    float*       out = (float*)d_out;

    float* ws    = (float*)d_ws;
    float* part  = ws;                              // 64 * 128 floats
    float* s     = ws + COLSUM_BLOCKS * D_DIM;      // 128 floats
    float* wsum  = s + D_DIM;                       // 4096 floats

    sg_colsum_partial<<<COLSUM_BLOCKS, D_DIM, 0, stream>>>(E, part);
    sg_colsum_final<<<1, D_DIM, 0, stream>>>(part, s);
    sg_pair_cost_wmma<<<WAVES_TOTAL / 8, 256, 0, stream>>>(E, iv, jv, s, wsum);
    sg_final_reduce<<<1, 256, 0, stream>>>(wsum, out);
}